// DNNTSP_43087111914237
// MI455X (gfx1250) — compile-verified
//
#include <hip/hip_runtime.h>

#define B_ 32
#define NG_ 64
#define T_ 12
#define F_ 64
#define ITEMS_ 30000
#define H_ 4
#define N_ (B_*NG_)
#define EPS_ 1e-5f
#define LSA 68   // stride (floats) for row-major A-side LDS tiles (272B = 16B aligned rows)
#define LSB 66   // stride (floats) for transposed B-side LDS tiles

typedef float v2f __attribute__((ext_vector_type(2)));
typedef float v8f __attribute__((ext_vector_type(8)));

#if defined(__gfx1250__) && __has_builtin(__builtin_amdgcn_global_load_async_to_lds_b128)
#define HAVE_ASYNC_LDS 1
#else
#define HAVE_ASYNC_LDS 0
#endif

#if HAVE_ASYNC_LDS
typedef int async_v4i __attribute__((vector_size(4 * sizeof(int))));
__device__ __forceinline__ void async_copy16(const float* gp, float* lp) {
  // Builtin expects (v4i __device__*, v4i __shared__*, imm offset, imm cpol).
  // Produce AS1/AS3 pointers via integer round-trip (generic shared address
  // low 32 bits == LDS byte offset).
  __builtin_amdgcn_global_load_async_to_lds_b128(
      (__attribute__((address_space(1))) async_v4i*)(unsigned long long)(const void*)gp,
      (__attribute__((address_space(3))) async_v4i*)(unsigned int)(unsigned long long)(void*)lp,
      0, 0);
}
__device__ __forceinline__ void async_wait0() {
#if __has_builtin(__builtin_amdgcn_s_wait_asynccnt)
  __builtin_amdgcn_s_wait_asynccnt(0);
#else
  asm volatile("s_wait_asynccnt 0" ::: "memory");
#endif
}
#endif

// One 16x16 f32 tile of a 64x64x64 GEMM. A row-major (stride LSA),
// B transposed in LDS (sBT[n][k], stride LSB) so both fragments are b64 loads.
// A frag: lane%16 = M, lane/16 selects K pair. B frag: lane%16 = N, same K pair.
__device__ __forceinline__ v8f wmma_tile(const float* __restrict__ sA,
                                         const float* __restrict__ sBT,
                                         int rowTile, int colTile, v8f c) {
  const int lane = threadIdx.x & 31;
  const int mn = lane & 15;
  const int hi = lane >> 4;
  const float* arow = sA + (rowTile * 16 + mn) * LSA;
  const float* brow = sBT + (colTile * 16 + mn) * LSB;
#pragma unroll
  for (int k = 0; k < 64; k += 4) {
    const int kk = k + 2 * hi;
    v2f a = *(const v2f*)(arow + kk);
    v2f b = *(const v2f*)(brow + kk);
    c = __builtin_amdgcn_wmma_f32_16x16x4_f32(false, a, false, b, (short)0, c,
                                              false, false);
  }
  return c;
}

__global__ void init_stats_kernel(float* __restrict__ stats) {
  int i = threadIdx.x;
  if (i < 256) stats[i] = 0.0f;
}

// One block per (b,t). hout[b,t] = (A[b,t] @ hin) @ W + bias, plus per-channel
// sum/sumsq atomics for BatchNorm. layer==2 applies BN+relu to input on load.
__global__ __launch_bounds__(512) void gcn_layer_kernel(
    const float* __restrict__ edges, const float* __restrict__ hin,
    const float* __restrict__ W, const float* __restrict__ bias,
    const float* __restrict__ bnscale, const float* __restrict__ bnshift,
    float* __restrict__ hout, float* __restrict__ ssum, float* __restrict__ ssq,
    int layer) {
  __shared__ float sA[64 * LSA];   // A matrix, then result staging for stats
  __shared__ float sM[64 * LSA];   // intermediate m = A @ h
  __shared__ float sBT[64 * LSB];  // transposed B operand (h, then W)
  const int tid = threadIdx.x;
  const int b = blockIdx.x / T_;
  const int t = blockIdx.x % T_;
  const float* Ab = edges + (size_t)t * (B_ * NG_ * NG_) + (size_t)b * (NG_ * NG_);

#if HAVE_ASYNC_LDS
  // A[b,t]: 64 rows x 256B, copy as 1024 16B chunks via async DMA to LDS.
#pragma unroll
  for (int it = 0; it < 2; ++it) {
    int cidx = tid + it * 512;
    int row = cidx >> 4, seg = cidx & 15;
    async_copy16(Ab + row * 64 + seg * 4, &sA[row * LSA + seg * 4]);
  }
#else
#pragma unroll
  for (int it = 0; it < 8; ++it) {
    int e = tid + it * 512;
    sA[(e >> 6) * LSA + (e & 63)] = Ab[e];
  }
#endif
  // h operand, stored transposed: sBT[f][j] = h[j][f]
#pragma unroll
  for (int it = 0; it < 8; ++it) {
    int e = tid + it * 512;  // e = j*64 + f
    int j = e >> 6, f = e & 63;
    float v;
    if (layer == 1) {
      v = hin[(size_t)(b * NG_ + j) * F_ + f];
    } else {
      v = hin[(((size_t)b * T_ + t) * NG_ + j) * F_ + f];
      v = fmaxf(v * bnscale[f] + bnshift[f], 0.0f);
    }
    sBT[f * LSB + j] = v;
  }
#if HAVE_ASYNC_LDS
  async_wait0();
#endif
  __syncthreads();

  const int wave = tid >> 5;
  const int wi = wave >> 2, wj = wave & 3;
  const int lane = tid & 31;
  const int nn = lane & 15, hi = lane >> 4;
  v8f c1 = {};
  c1 = wmma_tile(sA, sBT, wi, wj, c1);  // m = A @ h
  __syncthreads();                       // stage-1 LDS reads complete
#pragma unroll
  for (int r = 0; r < 8; ++r)
    sM[(wi * 16 + r + 8 * hi) * LSA + wj * 16 + nn] = c1[r];
#pragma unroll
  for (int it = 0; it < 8; ++it) {  // W transposed into sBT
    int e = tid + it * 512;         // e = k*64 + n
    sBT[(e & 63) * LSB + (e >> 6)] = W[e];
  }
  __syncthreads();
  v8f c2 = {};
  c2 = wmma_tile(sM, sBT, wi, wj, c2);  // hpre = m @ W
  const float bb = bias[wj * 16 + nn];
#pragma unroll
  for (int r = 0; r < 8; ++r) {
    float v = c2[r] + bb;
    int row = wi * 16 + r + 8 * hi;
    int col = wj * 16 + nn;
    hout[(((size_t)b * T_ + t) * NG_ + row) * F_ + col] = v;
    sA[row * LSA + col] = v;  // sA free after stage-1 barrier
  }
  __syncthreads();
  if (tid < 64) {
    float s = 0.f, q = 0.f;
#pragma unroll
    for (int i = 0; i < 64; ++i) {
      float v = sA[i * LSA + tid];
      s += v;
      q += v * v;
    }
    atomicAdd(&ssum[tid], s);
    atomicAdd(&ssq[tid], q);
  }
}

__global__ void bn_finalize_kernel(const float* __restrict__ ssum,
                                   const float* __restrict__ ssq,
                                   const float* __restrict__ gamma,
                                   const float* __restrict__ beta,
                                   float* __restrict__ scale,
                                   float* __restrict__ shift) {
  int f = threadIdx.x;
  if (f < 64) {
    const float cnt = (float)(N_ * T_);
    float m = ssum[f] / cnt;
    float v = ssq[f] / cnt - m * m;
    float inv = rsqrtf(v + EPS_);
    float sc = gamma[f] * inv;
    scale[f] = sc;
    shift[f] = beta[f] - m * sc;
  }
}

// One block (4 waves) per node n = b*64+i. Q/K/V via WMMA (wave h == head h,
// head h occupies feature columns 16h..16h+15), tiny causal softmax in VALU,
// ragged linear aggregation, then aggdot[n] = agg . fc_W.
__global__ __launch_bounds__(128) void attn_agg_kernel(
    const float* __restrict__ hbuf, const float* __restrict__ bnscale,
    const float* __restrict__ bnshift, const float* __restrict__ Wq,
    const float* __restrict__ Wk, const float* __restrict__ Wv,
    const float* __restrict__ aggWq, const float* __restrict__ fcW,
    const int* __restrict__ lengths, float* __restrict__ aggdot) {
  __shared__ float sH[16 * LSB];       // h2 rows (A operand), rows 12..15 zero
  __shared__ float sWqT[64 * LSB];     // transposed weights (B operands)
  __shared__ float sWkT[64 * LSB];
  __shared__ float sWvT[64 * LSB];
  __shared__ float sQ[H_ * 16 * 17];
  __shared__ float sK[H_ * 16 * 17];
  __shared__ float sV[H_ * 16 * 17];
  __shared__ float sO[16 * LSB];
  __shared__ float sw[16];
  __shared__ float sred[64];
  const int tid = threadIdx.x;
  const int n = blockIdx.x;
  const int b = n >> 6;
  const int i = n & 63;
#pragma unroll
  for (int it = 0; it < 8; ++it) {
    int e = tid + it * 128;  // 16 rows x 64
    int t = e >> 6, f = e & 63;
    float v = 0.0f;
    if (t < T_) {
      v = hbuf[(((size_t)b * T_ + t) * NG_ + i) * F_ + f];
      v = fmaxf(v * bnscale[f] + bnshift[f], 0.0f);
    }
    sH[t * LSB + f] = v;
  }
  for (int e = tid; e < 4096; e += 128) {  // e = k*64 + n  (coalesced reads)
    int k = e >> 6, c = e & 63;
    sWqT[c * LSB + k] = Wq[e];
    sWkT[c * LSB + k] = Wk[e];
    sWvT[c * LSB + k] = Wv[e];
  }
  __syncthreads();
  const int h = tid >> 5;  // head == wave
  const int lane = tid & 31;
  const int nn = lane & 15, hi = lane >> 4;
  v8f cq = {}, ck = {}, cv = {};
  {
    const float* arow = sH + nn * LSB;
    const float* bq_row = sWqT + (16 * h + nn) * LSB;
    const float* bk_row = sWkT + (16 * h + nn) * LSB;
    const float* bv_row = sWvT + (16 * h + nn) * LSB;
#pragma unroll
    for (int k = 0; k < 64; k += 4) {
      int kk = k + 2 * hi;
      v2f a = *(const v2f*)(arow + kk);
      v2f bq = *(const v2f*)(bq_row + kk);
      v2f bk = *(const v2f*)(bk_row + kk);
      v2f bv = *(const v2f*)(bv_row + kk);
      cq = __builtin_amdgcn_wmma_f32_16x16x4_f32(false, a, false, bq, (short)0,
                                                 cq, false, false);
      ck = __builtin_amdgcn_wmma_f32_16x16x4_f32(false, a, false, bk, (short)0,
                                                 ck, false, false);
      cv = __builtin_amdgcn_wmma_f32_16x16x4_f32(false, a, false, bv, (short)0,
                                                 cv, false, false);
    }
  }
#pragma unroll
  for (int r = 0; r < 8; ++r) {
    int row = r + 8 * hi;
    sQ[(h * 16 + row) * 17 + nn] = cq[r];
    sK[(h * 16 + row) * 17 + nn] = ck[r];
    sV[(h * 16 + row) * 17 + nn] = cv[r];
  }
  __syncthreads();
  // causal softmax + att_out; lanes 0..11 of each wave own one time row
  if (lane < T_) {
    const int t = lane;
    float sc[T_];
    float mx = -3.0e38f;
    for (int s = 0; s <= t; ++s) {
      float d = 0.f;
#pragma unroll
      for (int dd = 0; dd < 16; ++dd)
        d += sQ[(h * 16 + t) * 17 + dd] * sK[(h * 16 + s) * 17 + dd];
      d *= 0.25f;  // 1/sqrt(DH), DH=16
      sc[s] = d;
      mx = fmaxf(mx, d);
    }
    float sum = 0.f;
    for (int s = 0; s <= t; ++s) {
      float e = __expf(sc[s] - mx);
      sc[s] = e;
      sum += e;
    }
    float inv = 1.0f / sum;
#pragma unroll
    for (int dd = 0; dd < 16; ++dd) {
      float o = 0.f;
      for (int s = 0; s <= t; ++s) o += sc[s] * sV[(h * 16 + s) * 17 + dd];
      sO[t * LSB + 16 * h + dd] = o * inv;
    }
  }
  __syncthreads();
  if (tid < T_) {
    const int t = tid;
    float wv = 0.f;
#pragma unroll
    for (int f = 0; f < 64; ++f) wv += sO[t * LSB + f] * aggWq[f];
    if (t >= lengths[b]) wv = 0.f;  // ragged mask
    sw[t] = wv;
  }
  __syncthreads();
  if (tid < 64) {
    float acc = 0.f;
#pragma unroll
    for (int t = 0; t < T_; ++t) acc += sw[t] * sO[t * LSB + tid];
    sred[tid] = acc * fcW[tid];
  }
  __syncthreads();
  if (tid == 0) {
    float s = 0.f;
#pragma unroll
    for (int f = 0; f < 64; ++f) s += sred[f];
    aggdot[n] = s;
  }
}

// base[i] = emb[i] . fcW ; out[b,i] = base[i] + fc_b  (broadcast over b)
__global__ void base_item_kernel(const float* __restrict__ emb,
                                 const float* __restrict__ fcW,
                                 const float* __restrict__ fcb,
                                 float* __restrict__ out,
                                 float* __restrict__ base) {
  int i = blockIdx.x * blockDim.x + threadIdx.x;
  if (i >= ITEMS_) return;
  float s = 0.f;
#pragma unroll
  for (int f = 0; f < 64; ++f) s += emb[(size_t)i * F_ + f] * fcW[f];
  base[i] = s;
  float o = s + fcb[0];
#pragma unroll
  for (int b = 0; b < B_; ++b) out[(size_t)b * ITEMS_ + i] = o;
}

// Gated corrections: out[b,item] += -alpha*base (once per distinct item) and
// alpha*aggdot (per occurrence; duplicates accumulate like scatter-add).
__global__ void scatter_kernel(const int* __restrict__ nodes,
                               const float* __restrict__ alpha,
                               const float* __restrict__ base,
                               const float* __restrict__ aggdot,
                               float* __restrict__ out) {
  int idx = blockIdx.x * blockDim.x + threadIdx.x;
  if (idx >= N_) return;
  int b = idx >> 6, g = idx & 63;
  int item = nodes[idx];
  float a = alpha[item];
  bool first = true;
  for (int gg = 0; gg < g; ++gg)
    if (nodes[b * NG_ + gg] == item) { first = false; break; }
  float delta = a * aggdot[idx] + (first ? -a * base[item] : 0.0f);
  atomicAdd(&out[(size_t)b * ITEMS_ + item], delta);
}

extern "C" void kernel_launch(void* const* d_in, const int* in_sizes, int n_in,
                              void* d_out, int out_size, void* d_ws,
                              size_t ws_size, hipStream_t stream) {
  const float* nodes_feature = (const float*)d_in[0];
  const float* edges_weight = (const float*)d_in[1];
  const int* lengths = (const int*)d_in[2];
  const int* nodes = (const int*)d_in[3];
  // d_in[4] users_frequency: unused by the reference forward
  const float* emb_table = (const float*)d_in[5];
  const float* W1 = (const float*)d_in[6];
  const float* b1 = (const float*)d_in[7];
  const float* g1 = (const float*)d_in[8];
  const float* be1 = (const float*)d_in[9];
  const float* W2 = (const float*)d_in[10];
  const float* b2 = (const float*)d_in[11];
  const float* g2 = (const float*)d_in[12];
  const float* be2 = (const float*)d_in[13];
  const float* Wq = (const float*)d_in[14];
  const float* Wk = (const float*)d_in[15];
  const float* Wv = (const float*)d_in[16];
  const float* aggWq = (const float*)d_in[17];
  const float* alpha = (const float*)d_in[18];
  const float* fcW = (const float*)d_in[19];
  const float* fcb = (const float*)d_in[20];
  float* out = (float*)d_out;

  float* ws = (float*)d_ws;
  float* hbuf = ws;                                  // B*T*64*64 = 1572864 f
  float* stats = hbuf + (size_t)B_ * T_ * NG_ * F_;  // 256 f
  float* sum1 = stats;
  float* sq1 = stats + 64;
  float* sum2 = stats + 128;
  float* sq2 = stats + 192;
  float* bnp = stats + 256;  // 256 f
  float* scale1 = bnp;
  float* shift1 = bnp + 64;
  float* scale2 = bnp + 128;
  float* shift2 = bnp + 192;
  float* aggd = bnp + 256;  // 2048 f
  float* base = aggd + N_;  // 30000 f

  init_stats_kernel<<<1, 256, 0, stream>>>(stats);

  gcn_layer_kernel<<<B_ * T_, 512, 0, stream>>>(
      edges_weight, nodes_feature, W1, b1, (const float*)nullptr,
      (const float*)nullptr, hbuf, sum1, sq1, 1);
  bn_finalize_kernel<<<1, 64, 0, stream>>>(sum1, sq1, g1, be1, scale1, shift1);

  gcn_layer_kernel<<<B_ * T_, 512, 0, stream>>>(
      edges_weight, hbuf, W2, b2, scale1, shift1, hbuf, sum2, sq2, 2);
  bn_finalize_kernel<<<1, 64, 0, stream>>>(sum2, sq2, g2, be2, scale2, shift2);

  attn_agg_kernel<<<N_, 128, 0, stream>>>(hbuf, scale2, shift2, Wq, Wk, Wv,
                                          aggWq, fcW, lengths, aggd);

  base_item_kernel<<<(ITEMS_ + 255) / 256, 256, 0, stream>>>(emb_table, fcW,
                                                             fcb, out, base);
  scatter_kernel<<<(N_ + 255) / 256, 256, 0, stream>>>(nodes, alpha, base,
                                                       aggd, out);
}